// HGT_22333829939350
// MI455X (gfx1250) — compile-verified
//
#include <hip/hip_runtime.h>
#include <math.h>

// Problem constants (match reference)
#define N_NODES 50000
#define D_FEAT  128
#define R_REL   3
#define E_EDGES 800000
#define OUT_FEAT 64
#define ALPHA 0.5f
#define SLOPE 0.01f

typedef __attribute__((ext_vector_type(2))) float v2f;
typedef __attribute__((ext_vector_type(8))) float v8f;

// ---------------------------------------------------------------- utilities
__global__ void zero_f4(float4* p, int n4) {
  int i = blockIdx.x * blockDim.x + threadIdx.x;
  if (i < n4) p[i] = make_float4(0.f, 0.f, 0.f, 0.f);
}

__global__ void count_deg(const int* __restrict__ src, const int* __restrict__ dst,
                          float* deg_out, float* deg_in, int E) {
  int e = blockIdx.x * blockDim.x + threadIdx.x;
  if (e >= E) return;
  atomicAdd(&deg_out[src[e]], 1.0f);
  atomicAdd(&deg_in[dst[e]], 1.0f);
}

__global__ void finish_norm(float* v, int n) {
  int i = blockIdx.x * blockDim.x + threadIdx.x;
  if (i < n) v[i] = rsqrtf(fmaxf(v[i], 1.0f));
}

// ------------------------------------------------- SpMM: one wave32 per edge
// agg[dst] += h[src] * rn_out[src]; each lane moves a float4 (512B/edge, coalesced)
__global__ void scatter_edges(const float* __restrict__ h, const int* __restrict__ src,
                              const int* __restrict__ dst, const float* __restrict__ rn_out,
                              float* __restrict__ agg, int E) {
  int idx = blockIdx.x * blockDim.x + threadIdx.x;
  int e = idx >> 5, lane = idx & 31;
  if (e >= E) return;
  int s = src[e], d = dst[e];
  float w = rn_out[s];
  float4 v = ((const float4*)(h + (size_t)s * D_FEAT))[lane];
  float* ap = agg + (size_t)d * D_FEAT + lane * 4;
  atomicAdd(ap + 0, v.x * w);
  atomicAdd(ap + 1, v.y * w);
  atomicAdd(ap + 2, v.z * w);
  atomicAdd(ap + 3, v.w * w);
}

// rst = (1-ALPHA) * agg * rn_in[node] + ALPHA * feat0   (float4 vectorized)
__global__ void make_rst(const float4* __restrict__ agg, const float* __restrict__ rn_in,
                         const float4* __restrict__ feat0, float4* __restrict__ rst, int n4) {
  int i = blockIdx.x * blockDim.x + threadIdx.x;
  if (i >= n4) return;
  int node = i >> 5;                       // 32 float4 per node (D=128)
  float w = rn_in[node] * (1.0f - ALPHA);
  float4 a = agg[i], f = feat0[i], o;
  o.x = a.x * w + ALPHA * f.x;
  o.y = a.y * w + ALPHA * f.y;
  o.z = a.z * w + ALPHA * f.z;
  o.w = a.w * w + ALPHA * f.w;
  rst[i] = o;
}

// ----------------------------------------------- fp32 WMMA GEMM (16x16x4_f32)
// A-frag (16x4 f32): lane<16 -> row=lane, k-pair {0,1}; lane>=16 -> row=lane-16, k-pair {2,3}
// C/D   (16x16 f32): VGPR j -> M = j + (lane>>4)*8, N = lane&15
#define AS_S 132   // padded LDS stride (128+4) -> conflict-free column reads

// hout += scale * act( (1-beta)*rst + beta*(rst @ W) + bias )
// 8 waves: each owns one 16-wide column tile of the 16x128 output block.
__global__ void __launch_bounds__(256)
gemm_combine(const float* __restrict__ rst, const float* __restrict__ W,
             const float* __restrict__ bias, float* __restrict__ hout,
             float beta, int act, float scale) {
  __shared__ float As[16 * AS_S];
  int m0 = blockIdx.x * 16;
  const float4* ap = (const float4*)(rst + (size_t)m0 * D_FEAT);
  for (int t = threadIdx.x; t < 16 * 32; t += 256) {
    int row = t >> 5, c4 = t & 31;
    ((float4*)(As + row * AS_S))[c4] = ap[t];
  }
  __syncthreads();

  int lane = threadIdx.x & 31;
  int n0 = (threadIdx.x >> 5) * 16;
  int rl = lane & 15;
  int kb = (lane >> 4) * 2;
  v8f c = {};
#pragma unroll 4
  for (int k = 0; k < D_FEAT; k += 4) {
    v2f a, b;
    a.x = As[rl * AS_S + k + kb];
    a.y = As[rl * AS_S + k + kb + 1];
    b.x = W[(size_t)(k + kb) * D_FEAT + n0 + rl];
    b.y = W[(size_t)(k + kb + 1) * D_FEAT + n0 + rl];
    c = __builtin_amdgcn_wmma_f32_16x16x4_f32(false, a, false, b, (short)0, c, false, false);
  }

  int col = n0 + rl;
  int rbase = (lane >> 4) * 8;
  float bc = bias[col];
  float omb = 1.0f - beta;
#pragma unroll
  for (int j = 0; j < 8; ++j) {
    int m = rbase + j;
    float rv = As[m * AS_S + col];              // residual term from LDS
    float v = omb * rv + beta * c[j] + bc;
    if (act) v = (v >= 0.f) ? v : SLOPE * v;
    hout[(size_t)(m0 + m) * D_FEAT + col] += v * scale;   // unique (m,col) per launch
  }
}

// out = h2 @ Wlin + blin   (OUT=64 -> 4 waves/block)
__global__ void __launch_bounds__(128)
gemm_final(const float* __restrict__ h, const float* __restrict__ Wlin,
           const float* __restrict__ blin, float* __restrict__ out) {
  __shared__ float As[16 * AS_S];
  int m0 = blockIdx.x * 16;
  const float4* ap = (const float4*)(h + (size_t)m0 * D_FEAT);
  for (int t = threadIdx.x; t < 16 * 32; t += 128) {
    int row = t >> 5, c4 = t & 31;
    ((float4*)(As + row * AS_S))[c4] = ap[t];
  }
  __syncthreads();

  int lane = threadIdx.x & 31;
  int n0 = (threadIdx.x >> 5) * 16;
  int rl = lane & 15;
  int kb = (lane >> 4) * 2;
  v8f c = {};
#pragma unroll 4
  for (int k = 0; k < D_FEAT; k += 4) {
    v2f a, b;
    a.x = As[rl * AS_S + k + kb];
    a.y = As[rl * AS_S + k + kb + 1];
    b.x = Wlin[(size_t)(k + kb) * OUT_FEAT + n0 + rl];
    b.y = Wlin[(size_t)(k + kb + 1) * OUT_FEAT + n0 + rl];
    c = __builtin_amdgcn_wmma_f32_16x16x4_f32(false, a, false, b, (short)0, c, false, false);
  }

  int col = n0 + rl;
  int rbase = (lane >> 4) * 8;
  float bc = blin[col];
#pragma unroll
  for (int j = 0; j < 8; ++j) {
    int m = rbase + j;
    out[(size_t)(m0 + m) * OUT_FEAT + col] = c[j] + bc;
  }
}

// ---------------------------------------------------------------- launcher
extern "C" void kernel_launch(void* const* d_in, const int* in_sizes, int n_in,
                              void* d_out, int out_size, void* d_ws, size_t ws_size,
                              hipStream_t stream) {
  const float* x    = (const float*)d_in[0];
  const int*   src  = (const int*)d_in[1];
  const int*   dst  = (const int*)d_in[2];
  const float* W1   = (const float*)d_in[3];
  const float* b1   = (const float*)d_in[4];
  const float* W2   = (const float*)d_in[5];
  const float* b2   = (const float*)d_in[6];
  const float* Wlin = (const float*)d_in[7];
  const float* blin = (const float*)d_in[8];
  float* out = (float*)d_out;

  // Workspace layout (~104 MB): norms + agg + rst + h1 + h2
  float* rn_out = (float*)d_ws;                       // R*N
  float* rn_in  = rn_out + R_REL * N_NODES;           // R*N
  float* agg    = rn_in  + R_REL * N_NODES;           // N*D
  float* rst    = agg + (size_t)N_NODES * D_FEAT;     // N*D
  float* h1     = rst + (size_t)N_NODES * D_FEAT;     // N*D
  float* h2     = h1  + (size_t)N_NODES * D_FEAT;     // N*D

  const float BETA1 = 0.69314718055994530942f;  // log(2)
  const float BETA2 = 0.40546510810816438198f;  // log(1.5)

  auto cdiv = [](int a, int b) { return (a + b - 1) / b; };

  // --- degrees -> rsqrt norms (shared by both layers) ---
  int degN = 2 * R_REL * N_NODES;
  zero_f4<<<cdiv(degN / 4, 256), 256, 0, stream>>>((float4*)rn_out, degN / 4);
  for (int r = 0; r < R_REL; ++r)
    count_deg<<<cdiv(E_EDGES, 256), 256, 0, stream>>>(
        src + (size_t)r * E_EDGES, dst + (size_t)r * E_EDGES,
        rn_out + r * N_NODES, rn_in + r * N_NODES, E_EDGES);
  finish_norm<<<cdiv(degN, 256), 256, 0, stream>>>(rn_out, degN);

  const int nd4 = N_NODES * D_FEAT / 4;

  auto layer = [&](const float* h, const float* feat0, const float* Wl, const float* bl,
                   float beta, int act, float* hout) {
    zero_f4<<<cdiv(nd4, 256), 256, 0, stream>>>((float4*)hout, nd4);
    for (int r = 0; r < R_REL; ++r) {
      zero_f4<<<cdiv(nd4, 256), 256, 0, stream>>>((float4*)agg, nd4);
      scatter_edges<<<cdiv(E_EDGES * 32, 256), 256, 0, stream>>>(
          h, src + (size_t)r * E_EDGES, dst + (size_t)r * E_EDGES,
          rn_out + r * N_NODES, agg, E_EDGES);
      make_rst<<<cdiv(nd4, 256), 256, 0, stream>>>(
          (const float4*)agg, rn_in + r * N_NODES, (const float4*)feat0,
          (float4*)rst, nd4);
      gemm_combine<<<N_NODES / 16, 256, 0, stream>>>(
          rst, Wl + (size_t)r * D_FEAT * D_FEAT, bl + r * D_FEAT,
          hout, beta, act, 1.0f / R_REL);
    }
  };

  layer(x,  x, W1, b1, BETA1, 1, h1);   // conv1: leaky_relu, mean over relations
  layer(h1, x, W2, b2, BETA2, 0, h2);   // conv2: no activation
  gemm_final<<<N_NODES / 16, 128, 0, stream>>>(h2, Wlin, blin, out);
}